// AirFitMultiHeadDNN_51084341019194
// MI455X (gfx1250) — compile-verified
//
#include <hip/hip_runtime.h>
#include <math.h>

typedef float v2f __attribute__((ext_vector_type(2)));
typedef float v8f __attribute__((ext_vector_type(8)));

#define H_        20
#define NUM_EX_   13
#define EMB_      3
#define FEAT_IN_  3
#define FEAT_OUT_ 5
#define HEAD_IN_  8
#define HEAD_HID_ 10

__global__ __launch_bounds__(256) void airfit_mh_kernel(
    const int*   __restrict__ e,    // (B,20) int32
    const float* __restrict__ f,    // (B,60)
    const float* __restrict__ emb,  // (13,3)
    const float* __restrict__ Wf,   // (3,5)
    const float* __restrict__ bf,   // (5)
    const float* __restrict__ W1,   // (20,8,10)
    const float* __restrict__ b1,   // (20,10)
    const float* __restrict__ W2,   // (20,10)
    const float* __restrict__ b2,   // (20)
    const float* __restrict__ Wo,   // (20)
    const float* __restrict__ bo,   // (1)
    float*       __restrict__ out,  // (B)
    int Btot)
{
    // Preformatted A operands: per head, 2 K-chunks, 32 lanes, float2 each.
    __shared__ __align__(16) v2f   sA[H_ * 2 * 32];     // 10240 B
    __shared__ __align__(16) float sB1p[H_ * 16];       // b1 zero-padded to 16 hid
    __shared__ __align__(16) float sW2p[H_ * 16];       // W2 zero-padded to 16 hid
    __shared__ float sEmb[NUM_EX_ * EMB_];
    __shared__ float sWf[FEAT_IN_ * FEAT_OUT_];
    __shared__ float sBf[FEAT_OUT_];
    __shared__ float sB2[H_];
    __shared__ float sWo[H_];
    __shared__ float sBo;

    const int tid = threadIdx.x;

    // ---- Preformat W1^T into the fp32 16x4 A-operand layout ----
    // i = h*128 + c*64 + lane*2 + comp
    // lane<16 holds K = c*4 + {0,1}; lane>=16 holds K = c*4 + {2,3}; M (=hid) = lane&15.
    float* sAf = (float*)sA;
    for (int i = tid; i < H_ * 2 * 32 * 2; i += blockDim.x) {
        int comp = i & 1;
        int lane = (i >> 1) & 31;
        int c    = (i >> 6) & 1;
        int h    = i >> 7;
        int m    = lane & 15;                              // hid row of A = W1^T
        int k    = c * 4 + ((lane & 16) ? 2 : 0) + comp;   // input index 0..7
        sAf[i] = (m < HEAD_HID_) ? W1[h * (HEAD_IN_ * HEAD_HID_) + k * HEAD_HID_ + m]
                                 : 0.0f;                   // zero-pad rows 10..15
    }
    // Zero-padded b1 / W2: unconditional, vectorizable LDS reads in the hot loop.
    for (int i = tid; i < H_ * 16; i += blockDim.x) {
        int h = i >> 4, hid = i & 15;
        float vb = (hid < HEAD_HID_) ? b1[h * HEAD_HID_ + hid] : 0.0f;
        float vw = (hid < HEAD_HID_) ? W2[h * HEAD_HID_ + hid] : 0.0f;
        sB1p[i] = vb;
        sW2p[i] = vw;
    }
    for (int i = tid; i < NUM_EX_ * EMB_; i += blockDim.x) sEmb[i] = emb[i];
    if (tid < FEAT_IN_ * FEAT_OUT_) sWf[tid] = Wf[tid];
    if (tid < FEAT_OUT_)            sBf[tid] = bf[tid];
    if (tid < H_)                   { sB2[tid] = b2[tid]; sWo[tid] = Wo[tid]; }
    if (tid == 0)                   sBo = bo[0];
    __syncthreads();

    const int lane   = tid & 31;
    const int loHalf = (lane < 16) ? 1 : 0;
    const int n      = lane & 15;                 // sample slot within the 16-wide tile
    const int wavesPerBlock = blockDim.x >> 5;
    const int waveGlobal    = blockIdx.x * wavesPerBlock + (tid >> 5);
    const int totalWaves    = gridDim.x * wavesPerBlock;
    const int numTiles      = (Btot + 15) >> 4;

    // Feature-linear columns this lane needs:
    // lanes 0-15 need fv1, fv2 ; lanes 16-31 need fv0, fv3, fv4.
    const int cA = loHalf ? 1 : 0;
    const int cB = loHalf ? 2 : 3;
    const int cC = loHalf ? 2 : 4;
    const int hbase = loHalf ? 0 : 8;

    for (int tile = waveGlobal; tile < numTiles; tile += totalWaves) {
        int  b      = tile * 16 + n;
        int  bc     = (b < Btot) ? b : (Btot - 1);   // clamp: keep EXEC uniform for WMMA
        const int*   eRow = e + bc * H_;
        const float* fRow = f + bc * (H_ * FEAT_IN_);
        float accOut = 0.0f;

        #pragma unroll 4
        for (int h = 0; h < H_; ++h) {
            // ---- build this lane's B-operand entries (x^T) ----
            int   idx = eRow[h];
            float f0 = fRow[h * 3 + 0];
            float f1 = fRow[h * 3 + 1];
            float f2 = fRow[h * 3 + 2];
            float ev0 = sEmb[idx * 3 + 0];
            float ev1 = sEmb[idx * 3 + 1];
            float ev2 = sEmb[idx * 3 + 2];
            float fvA = sBf[cA] + f0 * sWf[0 * 5 + cA] + f1 * sWf[1 * 5 + cA] + f2 * sWf[2 * 5 + cA];
            float fvB = sBf[cB] + f0 * sWf[0 * 5 + cB] + f1 * sWf[1 * 5 + cB] + f2 * sWf[2 * 5 + cB];
            float fvC = sBf[cC] + f0 * sWf[0 * 5 + cC] + f1 * sWf[1 * 5 + cC] + f2 * sWf[2 * 5 + cC];
            // x = [ev0 ev1 ev2 fv0 fv1 fv2 fv3 fv4]; lane<16 supplies K {0,1,4,5}, lane>=16 K {2,3,6,7}
            v2f bv0, bv1;
            bv0.x = loHalf ? ev0 : ev2;
            bv0.y = loHalf ? ev1 : fvA;
            bv1.x = loHalf ? fvA : fvB;
            bv1.y = loHalf ? fvB : fvC;

            v2f a0 = sA[(h * 2 + 0) * 32 + lane];
            v2f a1 = sA[(h * 2 + 1) * 32 + lane];

            // D(16x16) = W1^T(16x8) * x^T(8x16): two K=4 fp32 WMMAs
            v8f acc = {0.0f, 0.0f, 0.0f, 0.0f, 0.0f, 0.0f, 0.0f, 0.0f};
            acc = __builtin_amdgcn_wmma_f32_16x16x4_f32(false, a0, false, bv0,
                                                        (short)0, acc, false, false);
            acc = __builtin_amdgcn_wmma_f32_16x16x4_f32(false, a1, false, bv1,
                                                        (short)0, acc, false, false);

            // ---- bias + leaky_relu + dot with W2 (hid lives along VGPRs) ----
            // Padded tables: rows hid>=10 are zero, and acc rows >=10 are zero too,
            // so everything is unconditional (no exec masking, b128 LDS loads).
            const float4* b1v = (const float4*)(sB1p + h * 16 + hbase);
            const float4* w2v = (const float4*)(sW2p + h * 16 + hbase);
            float4 bb0 = b1v[0], bb1 = b1v[1];
            float4 ww0 = w2v[0], ww1 = w2v[1];

            float t, partial;
            t = acc[0] + bb0.x; t = (t > 0.0f) ? t : 0.01f * t; partial  = t * ww0.x;
            t = acc[1] + bb0.y; t = (t > 0.0f) ? t : 0.01f * t; partial += t * ww0.y;
            t = acc[2] + bb0.z; t = (t > 0.0f) ? t : 0.01f * t; partial += t * ww0.z;
            t = acc[3] + bb0.w; t = (t > 0.0f) ? t : 0.01f * t; partial += t * ww0.w;
            t = acc[4] + bb1.x; t = (t > 0.0f) ? t : 0.01f * t; partial += t * ww1.x;
            t = acc[5] + bb1.y; t = (t > 0.0f) ? t : 0.01f * t; partial += t * ww1.y;
            t = acc[6] + bb1.z; t = (t > 0.0f) ? t : 0.01f * t; partial += t * ww1.z;
            t = acc[7] + bb1.w; t = (t > 0.0f) ? t : 0.01f * t; partial += t * ww1.w;

            // combine lane-pair halves (hid 0-7 in lanes 0-15, hid 8-9 in lanes 16-31)
            partial += __shfl_xor(partial, 16, 32);

            float v  = partial + sB2[h];
            // softplus = max(v,0) + log1p(exp(-|v|)); fast hw exp/log, ~1e-7 abs err
            float sp = fmaxf(v, 0.0f) + __logf(1.0f + __expf(-fabsf(v)));
            accOut += sp * sWo[h];
        }

        if (loHalf && b < Btot) out[b] = accOut + sBo;
    }
}

extern "C" void kernel_launch(void* const* d_in, const int* in_sizes, int n_in,
                              void* d_out, int out_size, void* d_ws, size_t ws_size,
                              hipStream_t stream) {
    const int*   e   = (const int*)  d_in[0];
    const float* f   = (const float*)d_in[1];
    const float* emb = (const float*)d_in[2];
    const float* Wf  = (const float*)d_in[3];
    const float* bf  = (const float*)d_in[4];
    const float* W1  = (const float*)d_in[5];
    const float* b1  = (const float*)d_in[6];
    const float* W2  = (const float*)d_in[7];
    const float* b2  = (const float*)d_in[8];
    const float* Wo  = (const float*)d_in[9];
    const float* bo  = (const float*)d_in[10];
    float* out = (float*)d_out;

    int Btot = in_sizes[0] / H_;                 // e is (B, 20)
    int numTiles = (Btot + 15) / 16;
    int block = 256;                              // 8 wave32 per block
    int wavesPerBlock = block / 32;
    int grid = 1024;                              // ~4 tiles per wave at B=524288
    int maxGrid = (numTiles + wavesPerBlock - 1) / wavesPerBlock;
    if (grid > maxGrid) grid = maxGrid;
    if (grid < 1) grid = 1;

    airfit_mh_kernel<<<grid, block, 0, stream>>>(e, f, emb, Wf, bf, W1, b1, W2, b2,
                                                 Wo, bo, out, Btot);
}